// DMRDeNoise_49572512530920
// MI455X (gfx1250) — compile-verified
//
#include <hip/hip_runtime.h>

#define BB 4
#define NN 4096
#define KNN 16
#define CH 24      // CONV_CH
#define LO 12      // L_OUT
#define OC 60      // OUT_CH = 24 + 3*12
#define KEEP 2048
#define CHUNK 1024 // knn score columns resident in LDS at a time

typedef float v2f __attribute__((ext_vector_type(2)));
typedef float v8f __attribute__((ext_vector_type(8)));

// ---------------------------------------------------------------------------
// trans: h = (relu?)(x @ W^T + b), also emit squared norms of h rows
// ---------------------------------------------------------------------------
__global__ void dmr_trans_kernel(const float* __restrict__ xin, int in_ch,
                                 const float* __restrict__ W,
                                 const float* __restrict__ bias,
                                 int do_relu,
                                 float* __restrict__ hout,
                                 float* __restrict__ sqn) {
  int t = blockIdx.x * blockDim.x + threadIdx.x;   // point id over B*N
  if (t >= BB * NN) return;
  const float* xp = xin + (size_t)t * in_ch;
  float acc[CH];
  for (int o = 0; o < CH; ++o) {
    float a = bias[o];
    const float* w = W + o * in_ch;
    for (int c = 0; c < in_ch; ++c) a += w[c] * xp[c];
    if (do_relu) a = fmaxf(a, 0.0f);
    acc[o] = a;
  }
  float s = 0.0f;
  float* hp = hout + (size_t)t * CH;
  for (int o = 0; o < CH; ++o) { hp[o] = acc[o]; s += acc[o] * acc[o]; }
  sqn[t] = s;
}

// ---------------------------------------------------------------------------
// KNN: one block = 16 query points of one batch. 8 waves compute 16x16 score
// tiles via v_wmma_f32_16x16x4_f32 into LDS; 16 threads/query keep top-16.
// Score = |p|^2 - 2 q.p  (same ordering as squared distance per row).
// Dynamic LDS: dist[16*CHUNK] floats; candidate区 aliased onto dist at the end.
// ---------------------------------------------------------------------------
__global__ void dmr_knn_kernel(const float* __restrict__ h,
                               const float* __restrict__ sqn,
                               int* __restrict__ knn_out) {
  extern __shared__ float lds[];
  float* dist = lds;                          // 16*CHUNK floats
  float* candd = lds;                         // alias (after last phase2)
  int*   candi = (int*)(lds + 16 * 16 * 16);  // 4096 ints after candd

  const int blk  = blockIdx.x;
  const int b    = blk / (NN / 16);
  const int q0   = (blk % (NN / 16)) * 16;
  const int tid  = threadIdx.x;
  const int lane = tid & 31;
  const int wave = tid >> 5;
  const int hl   = (lane < 16) ? 0 : 2;   // A/B fragment K offset per half-wave
  const int mrow = lane & 15;

  const float* hb = h + (size_t)b * NN * CH;
  const float* sb = sqn + (size_t)b * NN;

  // A fragment: queries, K padded 24 -> 28 (col 24 = 1.0 for the |p|^2 term)
  v2f afrag[7];
  {
    const float* qrow = hb + (size_t)(q0 + mrow) * CH;
    #pragma unroll
    for (int c = 0; c < 7; ++c) {
      int k0 = 4 * c + hl, k1 = k0 + 1;
      afrag[c][0] = (k0 < CH) ? qrow[k0] : (k0 == CH ? 1.0f : 0.0f);
      afrag[c][1] = (k1 < CH) ? qrow[k1] : (k1 == CH ? 1.0f : 0.0f);
    }
  }

  float bestd[KNN]; int besti[KNN];
  #pragma unroll
  for (int j = 0; j < KNN; ++j) { bestd[j] = 3.4e38f; besti[j] = -1; }
  const int s  = tid & 15;   // sub-slot within query
  const int m  = tid >> 4;   // query row 0..15
  const int gq = q0 + m;

  for (int chunk = 0; chunk < NN / CHUNK; ++chunk) {
    // phase 1: fill dist[16][CHUNK] with WMMA tiles (64 tiles, 8 per wave)
    for (int i = 0; i < (CHUNK / 16) / 8; ++i) {
      int tt = wave + 8 * i;
      int n0 = chunk * CHUNK + tt * 16;
      const float* prow = hb + (size_t)(n0 + mrow) * CH;
      float pn2 = sb[n0 + mrow];
      v8f acc = {};
      #pragma unroll
      for (int c = 0; c < 7; ++c) {
        int k0 = 4 * c + hl, k1 = k0 + 1;
        v2f bf;
        bf[0] = (k0 < CH) ? -2.0f * prow[k0] : (k0 == CH ? pn2 : 0.0f);
        bf[1] = (k1 < CH) ? -2.0f * prow[k1] : (k1 == CH ? pn2 : 0.0f);
        acc = __builtin_amdgcn_wmma_f32_16x16x4_f32(
            false, afrag[c], false, bf, (short)0, acc, false, false);
      }
      int row0 = (lane < 16) ? 0 : 8;
      int col  = tt * 16 + mrow;
      #pragma unroll
      for (int v = 0; v < 8; ++v)
        dist[(row0 + v) * CHUNK + col] = acc[v];
    }
    __syncthreads();
    // phase 2: per (query, slot) running top-16 with self exclusion
    for (int i = 0; i < CHUNK / 16; ++i) {
      int lc = s + 16 * i;
      int gc = chunk * CHUNK + lc;
      if (gc == gq) continue;
      float d = dist[m * CHUNK + lc];
      if (d < bestd[KNN - 1]) {
        int j = KNN - 1;
        while (j > 0 && bestd[j - 1] > d) {
          bestd[j] = bestd[j - 1]; besti[j] = besti[j - 1]; --j;
        }
        bestd[j] = d; besti[j] = gc;
      }
    }
    __syncthreads();  // also protects the candidate-region alias below
  }

  // dump sorted per-slot candidates (aliases dist region — dist is dead now)
  #pragma unroll
  for (int j = 0; j < KNN; ++j) {
    candd[(m * 16 + s) * KNN + j] = bestd[j];
    candi[(m * 16 + s) * KNN + j] = besti[j];
  }
  __syncthreads();

  if (tid < 16) {
    int mm = tid;
    float fd[KNN]; int fi[KNN];
    #pragma unroll
    for (int j = 0; j < KNN; ++j) { fd[j] = 3.4e38f; fi[j] = -1; }
    for (int s2 = 0; s2 < 16; ++s2) {
      for (int j = 0; j < KNN; ++j) {
        float d = candd[(mm * 16 + s2) * KNN + j];
        if (d >= fd[KNN - 1]) break;          // lists are sorted ascending
        int id = candi[(mm * 16 + s2) * KNN + j];
        int p = KNN - 1;
        while (p > 0 && fd[p - 1] > d) { fd[p] = fd[p - 1]; fi[p] = fi[p - 1]; --p; }
        fd[p] = d; fi[p] = id;
      }
    }
    int* op = knn_out + ((size_t)b * NN + (q0 + mm)) * KNN;
    for (int j = 0; j < KNN; ++j) op[j] = fi[j];
  }
}

// ---------------------------------------------------------------------------
// DenseEdgeConv: block = 4 points (4 waves); each wave runs three WMMA GEMMs
// over its 16-edge tile; concat layout: [last(12) | mid(12) | first(12) | xt(24)]
// then max over the 16 edges.
// ---------------------------------------------------------------------------
__global__ void dmr_edgeconv_kernel(const float* __restrict__ h,
                                    const int* __restrict__ knn,
                                    const float* __restrict__ W1, const float* __restrict__ b1,
                                    const float* __restrict__ Wm, const float* __restrict__ bm,
                                    const float* __restrict__ Wl, const float* __restrict__ bl,
                                    float* __restrict__ xout) {
  __shared__ float sW1[LO * 72], sWm[LO * 36], sWl[LO * 48];
  __shared__ float sB1[LO], sBm[LO], sBl[LO];
  __shared__ float sYa[4][16 * 72];   // edge features; reused as final Yd (16x60)
  __shared__ float sYb[4][16 * 36];
  __shared__ float sYc[4][16 * 48];
  __shared__ float sXt[4][CH];

  const int tid  = threadIdx.x;
  const int lane = tid & 31;
  const int wave = tid >> 5;

  for (int t = tid; t < LO * 72; t += 128) sW1[t] = W1[t];
  for (int t = tid; t < LO * 36; t += 128) sWm[t] = Wm[t];
  for (int t = tid; t < LO * 48; t += 128) sWl[t] = Wl[t];
  if (tid < LO) { sB1[tid] = b1[tid]; sBm[tid] = bm[tid]; sBl[tid] = bl[tid]; }

  const int gpt = blockIdx.x * 4 + wave;     // global point id
  const int b   = gpt / NN;
  const int pt  = gpt % NN;
  const float* hb = h + (size_t)b * NN * CH;
  const int*   nb = knn + (size_t)gpt * KNN;

  if (lane < CH) sXt[wave][lane] = hb[(size_t)pt * CH + lane];
  __syncthreads();

  // edge = [x_t | x_j | x_j - x_t]  (16 x 72)
  for (int t = lane; t < 16 * CH; t += 32) {
    int e = t / CH, c = t % CH;
    float xj = hb[(size_t)nb[e] * CH + c];
    float xt = sXt[wave][c];
    sYa[wave][e * 72 + c]       = xt;
    sYa[wave][e * 72 + 24 + c]  = xj;
    sYa[wave][e * 72 + 48 + c]  = xj - xt;
  }
  __syncthreads();

  const int hl   = (lane < 16) ? 0 : 2;
  const int mrow = lane & 15;
  const int row0 = (lane < 16) ? 0 : 8;
  const int ncol = mrow;

  // layer "first": (16x72)@(72x12) -> relu -> Yb cols [0,12)
  {
    v8f acc = {};
    for (int c = 0; c < 18; ++c) {
      int k0 = 4 * c + hl;
      v2f af, bf;
      af[0] = sYa[wave][mrow * 72 + k0];
      af[1] = sYa[wave][mrow * 72 + k0 + 1];
      bf[0] = (ncol < LO) ? sW1[ncol * 72 + k0]     : 0.0f;
      bf[1] = (ncol < LO) ? sW1[ncol * 72 + k0 + 1] : 0.0f;
      acc = __builtin_amdgcn_wmma_f32_16x16x4_f32(
          false, af, false, bf, (short)0, acc, false, false);
    }
    if (ncol < LO) {
      #pragma unroll
      for (int v = 0; v < 8; ++v)
        sYb[wave][(row0 + v) * 36 + ncol] = fmaxf(acc[v] + sB1[ncol], 0.0f);
    }
  }
  for (int t = lane; t < 16 * CH; t += 32) {   // concat x_t into cols [12,36)
    int e = t / CH, c = t % CH;
    sYb[wave][e * 36 + 12 + c] = sXt[wave][c];
  }
  __syncthreads();

  // layer "mid": (16x36)@(36x12) -> relu -> Yc cols [0,12); copy prev to [12,48)
  {
    v8f acc = {};
    for (int c = 0; c < 9; ++c) {
      int k0 = 4 * c + hl;
      v2f af, bf;
      af[0] = sYb[wave][mrow * 36 + k0];
      af[1] = sYb[wave][mrow * 36 + k0 + 1];
      bf[0] = (ncol < LO) ? sWm[ncol * 36 + k0]     : 0.0f;
      bf[1] = (ncol < LO) ? sWm[ncol * 36 + k0 + 1] : 0.0f;
      acc = __builtin_amdgcn_wmma_f32_16x16x4_f32(
          false, af, false, bf, (short)0, acc, false, false);
    }
    if (ncol < LO) {
      #pragma unroll
      for (int v = 0; v < 8; ++v)
        sYc[wave][(row0 + v) * 48 + ncol] = fmaxf(acc[v] + sBm[ncol], 0.0f);
    }
  }
  for (int t = lane; t < 16 * 36; t += 32) {
    int e = t / 36, c = t % 36;
    sYc[wave][e * 48 + 12 + c] = sYb[wave][e * 36 + c];
  }
  __syncthreads();

  // layer "last": (16x48)@(48x12) -> Yd cols [0,12) (no relu); copy to [12,60)
  float* Yd = sYa[wave];   // alias: sYa no longer needed
  {
    v8f acc = {};
    for (int c = 0; c < 12; ++c) {
      int k0 = 4 * c + hl;
      v2f af, bf;
      af[0] = sYc[wave][mrow * 48 + k0];
      af[1] = sYc[wave][mrow * 48 + k0 + 1];
      bf[0] = (ncol < LO) ? sWl[ncol * 48 + k0]     : 0.0f;
      bf[1] = (ncol < LO) ? sWl[ncol * 48 + k0 + 1] : 0.0f;
      acc = __builtin_amdgcn_wmma_f32_16x16x4_f32(
          false, af, false, bf, (short)0, acc, false, false);
    }
    if (ncol < LO) {
      #pragma unroll
      for (int v = 0; v < 8; ++v)
        Yd[(row0 + v) * 60 + ncol] = acc[v] + sBl[ncol];
    }
  }
  for (int t = lane; t < 16 * 48; t += 32) {
    int e = t / 48, c = t % 48;
    Yd[e * 60 + 12 + c] = sYc[wave][e * 48 + c];
  }
  __syncthreads();

  // max over the 16 edges
  for (int c = lane; c < OC; c += 32) {
    float mx = -3.4e38f;
    for (int e = 0; e < 16; ++e) mx = fmaxf(mx, Yd[e * 60 + c]);
    xout[(size_t)gpt * OC + c] = mx;
  }
}

// ---------------------------------------------------------------------------
// GPool score
// ---------------------------------------------------------------------------
__global__ void dmr_score_kernel(const float* __restrict__ x,
                                 const float* __restrict__ pw,
                                 const float* __restrict__ pb,
                                 float* __restrict__ score) {
  int t = blockIdx.x * blockDim.x + threadIdx.x;
  if (t >= BB * NN) return;
  float nrm = 0.0f;
  for (int c = 0; c < OC; ++c) nrm += pw[c] * pw[c];
  nrm = sqrtf(nrm);
  float a = pb[0];
  const float* xp = x + (size_t)t * OC;
  for (int c = 0; c < OC; ++c) a += xp[c] * pw[c];
  score[t] = a / nrm;
}

// ---------------------------------------------------------------------------
// exact top-k(2048) by rank counting (ties broken by lower index, like top_k)
// ---------------------------------------------------------------------------
__global__ void dmr_topk_kernel(const float* __restrict__ score,
                                int* __restrict__ topidx,
                                float* __restrict__ topval,
                                float* __restrict__ out_idx_f) {
  __shared__ float s[NN];
  int b = blockIdx.x;
  const float* sc = score + (size_t)b * NN;
  for (int t = threadIdx.x; t < NN; t += blockDim.x) s[t] = sc[t];
  __syncthreads();
  for (int t = threadIdx.x; t < NN; t += blockDim.x) {
    float v = s[t];
    int r = 0;
    for (int j = 0; j < NN; ++j) {
      float u = s[j];
      r += (u > v) || (u == v && j < t);
    }
    if (r < KEEP) {
      topidx[b * KEEP + r]    = t;
      topval[b * KEEP + r]    = v;
      out_idx_f[b * KEEP + r] = (float)t;
    }
  }
}

// ---------------------------------------------------------------------------
// downsample + sigmoid gate + pre_filter MLP
// ---------------------------------------------------------------------------
__global__ void dmr_final_kernel(const float* __restrict__ x,
                                 const float* __restrict__ pos,
                                 const int* __restrict__ topidx,
                                 const float* __restrict__ topval,
                                 const float* __restrict__ M0W, const float* __restrict__ M0b,
                                 const float* __restrict__ M1W, const float* __restrict__ M1b,
                                 const float* __restrict__ M2W, const float* __restrict__ M2b,
                                 float* __restrict__ out_pos,
                                 float* __restrict__ out_xds) {
  int t = blockIdx.x * blockDim.x + threadIdx.x;
  if (t >= BB * KEEP) return;
  int b = t / KEEP;
  int n = topidx[t];
  float y = 1.0f / (1.0f + expf(-topval[t]));
  float xd[OC];
  const float* xp = x + ((size_t)b * NN + n) * OC;
  for (int c = 0; c < OC; ++c) xd[c] = xp[c] * y;
  float* od = out_xds + (size_t)t * OC;
  for (int c = 0; c < OC; ++c) od[c] = xd[c];
  float h1[30];
  for (int o = 0; o < 30; ++o) {
    float a = M0b[o];
    for (int c = 0; c < OC; ++c) a += M0W[o * OC + c] * xd[c];
    h1[o] = fmaxf(a, 0.0f);
  }
  float h2[15];
  for (int o = 0; o < 15; ++o) {
    float a = M1b[o];
    for (int c = 0; c < 30; ++c) a += M1W[o * 30 + c] * h1[c];
    h2[o] = fmaxf(a, 0.0f);
  }
  const float* pp = pos + ((size_t)b * NN + n) * 3;
  for (int o = 0; o < 3; ++o) {
    float a = M2b[o];
    for (int c = 0; c < 15; ++c) a += M2W[o * 15 + c] * h2[c];
    out_pos[(size_t)t * 3 + o] = pp[o] + a;
  }
}

// ---------------------------------------------------------------------------
extern "C" void kernel_launch(void* const* d_in, const int* in_sizes, int n_in,
                              void* d_out, int out_size, void* d_ws, size_t ws_size,
                              hipStream_t stream) {
  (void)out_size; (void)ws_size;
  // jax tree-flatten (sorted dict keys): convs, mlp, pool, trans, then pos.
  const float *cW1[4], *cb1[4], *cWl[4], *cbl[4], *cWm[4], *cbm[4];
  for (int i = 0; i < 4; ++i) {
    cW1[i] = (const float*)d_in[6 * i + 0];  // first.W (12x72)
    cb1[i] = (const float*)d_in[6 * i + 1];  // first.b
    cWl[i] = (const float*)d_in[6 * i + 2];  // last.W  (12x48)
    cbl[i] = (const float*)d_in[6 * i + 3];  // last.b
    cWm[i] = (const float*)d_in[6 * i + 4];  // mid.W   (12x36)
    cbm[i] = (const float*)d_in[6 * i + 5];  // mid.b
  }
  const float* m0W = (const float*)d_in[24];
  const float* m0b = (const float*)d_in[25];
  const float* m1W = (const float*)d_in[26];
  const float* m1b = (const float*)d_in[27];
  const float* m2W = (const float*)d_in[28];
  const float* m2b = (const float*)d_in[29];
  const float* poolW = (const float*)d_in[30];
  const float* poolb = (const float*)d_in[31];
  const float *trW[4], *trb[4];
  for (int i = 0; i < 4; ++i) {
    trW[i] = (const float*)d_in[32 + 2 * i];
    trb[i] = (const float*)d_in[33 + 2 * i];
  }
  int pos_i = n_in - 1;
  for (int i = 0; i < n_in; ++i)
    if (in_sizes[i] == BB * NN * 3) { pos_i = i; break; }
  const float* pos = (const float*)d_in[pos_i];

  // workspace layout
  float* ws    = (float*)d_ws;
  float* h     = ws;                                  // B*N*24
  float* sqn   = h + (size_t)BB * NN * CH;            // B*N
  float* xfeat = sqn + (size_t)BB * NN;               // B*N*60
  int*   knn   = (int*)(xfeat + (size_t)BB * NN * OC);// B*N*16
  float* score = (float*)(knn + (size_t)BB * NN * KNN);
  int*   topix = (int*)(score + (size_t)BB * NN);
  float* topvl = (float*)(topix + (size_t)BB * KEEP);

  // output layout: [top_idx(f32) | pos_out | x_ds]
  float* out_idx = (float*)d_out;
  float* out_pos = out_idx + (size_t)BB * KEEP;
  float* out_xds = out_pos + (size_t)BB * KEEP * 3;

  const int npts = BB * NN;
  const unsigned knn_lds = 16 * CHUNK * sizeof(float);  // 64 KB (cands alias it)

  for (int i = 0; i < 4; ++i) {
    const float* xin = (i == 0) ? pos : xfeat;
    int in_ch  = (i == 0) ? 3 : OC;
    int relu   = (i > 0) ? 1 : 0;
    dmr_trans_kernel<<<(npts + 255) / 256, 256, 0, stream>>>(
        xin, in_ch, trW[i], trb[i], relu, h, sqn);
    dmr_knn_kernel<<<BB * (NN / 16), 256, knn_lds, stream>>>(h, sqn, knn);
    dmr_edgeconv_kernel<<<npts / 4, 128, 0, stream>>>(
        h, knn, cW1[i], cb1[i], cWm[i], cbm[i], cWl[i], cbl[i], xfeat);
  }
  dmr_score_kernel<<<(npts + 255) / 256, 256, 0, stream>>>(xfeat, poolW, poolb, score);
  dmr_topk_kernel<<<BB, 256, 0, stream>>>(score, topix, topvl, out_idx);
  dmr_final_kernel<<<(BB * KEEP + 255) / 256, 256, 0, stream>>>(
      xfeat, pos, topix, topvl, m0W, m0b, m1W, m1b, m2W, m2b, out_pos, out_xds);
}